// GCNEncoder_39608188403884
// MI455X (gfx1250) — compile-verified
//
#include <hip/hip_runtime.h>

typedef float v2f __attribute__((ext_vector_type(2)));
typedef float v8f __attribute__((ext_vector_type(8)));

#define GDIN  128
#define GDH   128
#define GDOUT 64

// ---------------- utility: zero a float buffer ----------------
__global__ void zero_f32_kernel(float* __restrict__ p, size_t n) {
  size_t i = (size_t)blockIdx.x * blockDim.x + threadIdx.x;
  size_t stride = (size_t)gridDim.x * blockDim.x;
  for (; i < n; i += stride) p[i] = 0.0f;
}

// ---------------- degree accumulation (f32 atomics, L2-resident) ----------------
__global__ void degree_kernel(const int* __restrict__ dst, float* __restrict__ deg, int E) {
  int i = blockIdx.x * blockDim.x + threadIdx.x;
  int stride = gridDim.x * blockDim.x;
  for (; i < E; i += stride) atomicAdd(&deg[dst[i]], 1.0f);
}

// dinv[i] = rsqrt(deg[i] + 1)   (+1 = self loop; always > 0)
__global__ void rsqrt_kernel(float* __restrict__ dinv, int N) {
  int i = blockIdx.x * blockDim.x + threadIdx.x;
  if (i < N) dinv[i] = rsqrtf(dinv[i] + 1.0f);
}

// ---------------- f32 WMMA GEMM: T[N x NCOLS] = X[N x 128] @ W[128 x NCOLS] ----------------
// One block stages a 16x128 row-tile of X in LDS; each wave owns one 16x16 C tile
// and chains 32 V_WMMA_F32_16X16X4_F32 over K=128.
template <int NCOLS>
__global__ void __launch_bounds__(256) gemm_wmma_f32(const float* __restrict__ X,
                                                     const float* __restrict__ W,
                                                     float* __restrict__ T) {
  __shared__ float xs[16 * 128];
  const int tid = threadIdx.x;
  const int wave = tid >> 5;
  const int lane = tid & 31;
  const int nwaves = blockDim.x >> 5;
  const size_t m0 = (size_t)blockIdx.x * 16;

  // 16 consecutive rows of X are contiguous (row stride 128)
  for (int i = tid; i < 16 * 128; i += blockDim.x) xs[i] = X[m0 * 128 + i];
  __syncthreads();

  const int m = lane & 15;            // A-fragment row
  const int kg = (lane >> 4) << 1;    // K sub-group: 0 or 2
  const int ntiles = NCOLS >> 4;

  for (int nt = wave; nt < ntiles; nt += nwaves) {
    const int n = (nt << 4) + (lane & 15);   // B-fragment column
    v8f acc = {};
#pragma unroll
    for (int k0 = 0; k0 < 128; k0 += 4) {
      const int kb = k0 + kg;
      v2f a, b;
      a.x = xs[m * 128 + kb];
      a.y = xs[m * 128 + kb + 1];
      b.x = W[(size_t)kb * NCOLS + n];
      b.y = W[(size_t)(kb + 1) * NCOLS + n];
      // D = A(16x4,f32) x B(4x16,f32) + C(16x16,f32)
      acc = __builtin_amdgcn_wmma_f32_16x16x4_f32(false, a, false, b, (short)0, acc,
                                                  false, false);
    }
    const size_t rowbase = m0 + (size_t)((lane >> 4) << 3);
    const size_t col = (size_t)(nt << 4) + (lane & 15);
#pragma unroll
    for (int r = 0; r < 8; ++r) T[(rowbase + r) * NCOLS + col] = acc[r];
  }
}

// ---------------- edge scatter: OUT[dst] += T[src] * dinv[src]*dinv[dst] ----------------
// One edge per wave32; lanes cover the D features (coalesced loads + f32 atomics into L2).
template <int D>
__global__ void scatter_edges(const int* __restrict__ src, const int* __restrict__ dst,
                              const float* __restrict__ dinv, const float* __restrict__ T,
                              float* __restrict__ OUT, int E) {
  const int lane = threadIdx.x & 31;
  int w = (blockIdx.x * blockDim.x + threadIdx.x) >> 5;
  const int nw = (gridDim.x * blockDim.x) >> 5;
  for (; w < E; w += nw) {
    const int s = src[w];
    const int d = dst[w];
    const float nrm = dinv[s] * dinv[d];
    const float* tp = T + (size_t)s * D;
    float* op = OUT + (size_t)d * D;
#pragma unroll
    for (int f = 0; f < D / 32; ++f)
      atomicAdd(&op[lane + 32 * f], tp[lane + 32 * f] * nrm);
  }
}

// h = relu(agg + t1 * dinv[i]^2 + b1)   (self-loop message + bias + relu fused), in-place on t1
__global__ void finalize_relu_kernel(float* __restrict__ t1, const float* __restrict__ agg,
                                     const float* __restrict__ dinv,
                                     const float* __restrict__ bias, size_t total) {
  size_t idx = (size_t)blockIdx.x * blockDim.x + threadIdx.x;
  size_t stride = (size_t)gridDim.x * blockDim.x;
  for (; idx < total; idx += stride) {
    const size_t i = idx >> 7;        // / 128
    const int j = (int)(idx & 127);
    const float di = dinv[i];
    float v = agg[idx] + t1[idx] * di * di + bias[j];
    t1[idx] = v > 0.0f ? v : 0.0f;
  }
}

// out += t2 * dinv[i]^2 + b2   (out already holds edge aggregation)
__global__ void finalize_out_kernel(float* __restrict__ out, const float* __restrict__ t2,
                                    const float* __restrict__ dinv,
                                    const float* __restrict__ bias, size_t total) {
  size_t idx = (size_t)blockIdx.x * blockDim.x + threadIdx.x;
  size_t stride = (size_t)gridDim.x * blockDim.x;
  for (; idx < total; idx += stride) {
    const size_t i = idx >> 6;        // / 64
    const int j = (int)(idx & 63);
    const float di = dinv[i];
    out[idx] += t2[idx] * di * di + bias[j];
  }
}

extern "C" void kernel_launch(void* const* d_in, const int* in_sizes, int n_in,
                              void* d_out, int out_size, void* d_ws, size_t ws_size,
                              hipStream_t stream) {
  const float* x  = (const float*)d_in[0];
  const int*   ei = (const int*)d_in[1];   // [2,E] flat: row0=src, row1=dst
  const float* W1 = (const float*)d_in[2];
  const float* b1 = (const float*)d_in[3];
  const float* W2 = (const float*)d_in[4];
  const float* b2 = (const float*)d_in[5];
  float* out = (float*)d_out;

  const int N = in_sizes[0] / GDIN;
  const int E = in_sizes[1] / 2;
  const int* src = ei;
  const int* dst = ei + E;

  // workspace layout: dinv | bufA (t1 -> h) | bufB (agg1 -> t2)
  float* dinv = (float*)d_ws;
  float* bufA = dinv + N;
  float* bufB = bufA + (size_t)N * GDH;

  const size_t nh  = (size_t)N * GDH;    // N*128
  const size_t no  = (size_t)N * GDOUT;  // N*64
  const int mtiles = (N + 15) / 16;      // 6250 (exact: N = 100000)

  // 1) deg accumulation -> dinv
  zero_f32_kernel<<<512, 256, 0, stream>>>(dinv, (size_t)N);
  degree_kernel<<<(E + 255) / 256, 256, 0, stream>>>(dst, dinv, E);
  rsqrt_kernel<<<(N + 255) / 256, 256, 0, stream>>>(dinv, N);

  // 2) layer 1: t1 = x @ W1 (WMMA), agg over edges, fused self-loop + bias + relu
  gemm_wmma_f32<GDH><<<mtiles, 256, 0, stream>>>(x, W1, bufA);
  zero_f32_kernel<<<8192, 256, 0, stream>>>(bufB, nh);
  scatter_edges<GDH><<<16384, 256, 0, stream>>>(src, dst, dinv, bufA, bufB, E);
  finalize_relu_kernel<<<8192, 256, 0, stream>>>(bufA, bufB, dinv, b1, nh);

  // 3) layer 2: t2 = h @ W2 (WMMA), agg over edges into d_out, fused self-loop + bias
  gemm_wmma_f32<GDOUT><<<mtiles, 256, 0, stream>>>(bufA, W2, bufB);
  zero_f32_kernel<<<8192, 256, 0, stream>>>(out, no);
  scatter_edges<GDOUT><<<16384, 256, 0, stream>>>(src, dst, dinv, bufB, out, E);
  finalize_out_kernel<<<8192, 256, 0, stream>>>(out, bufB, dinv, b2, no);
}